// SOGPotential_63084479644017
// MI455X (gfx1250) — compile-verified
//
#include <hip/hip_runtime.h>

typedef __attribute__((ext_vector_type(2))) float v2f;
typedef __attribute__((ext_vector_type(8))) float v8f;

#define N_ATOMS 2000
#define NKGRID  33
#define KTOT    (NKGRID * NKGRID * NKGRID)   // 35937
#define BW      12
#define ATILES  (N_ATOMS / 16)               // 125 atom tiles
#define KTILES  ((KTOT + 15) / 16)           // 2247 k tiles
#define WAVES_PER_BLOCK 8
#define TPB     (WAVES_PER_BLOCK * 32)

__global__ void sog_init_kernel(float* out, const float* ene) {
    out[0] = ene[0];
}

__global__ __launch_bounds__(TPB)
void sog_main_kernel(const float* __restrict__ pos,
                     const float* __restrict__ q,
                     const float* __restrict__ cell,
                     const float* __restrict__ shift,
                     const float* __restrict__ amp,
                     float* __restrict__ out) {
    // A-operand fragments pre-swizzled into WMMA register order:
    //   aLDS[atom]           = (fx, fy)  -> K=0,1 rows (lanes 0-15, VGPR0/VGPR1)
    //   aLDS[N_ATOMS + atom] = (fz, 0)   -> K=2,3 rows (lanes 16-31)
    __shared__ v2f   aLDS[2 * N_ATOMS];
    __shared__ float qLDS[N_ATOMS];

    // ---- 3x3 inverse of cell (row-major), computed redundantly per thread ----
    float c00 = cell[0], c01 = cell[1], c02 = cell[2];
    float c10 = cell[3], c11 = cell[4], c12 = cell[5];
    float c20 = cell[6], c21 = cell[7], c22 = cell[8];
    float det = c00 * (c11 * c22 - c12 * c21)
              - c01 * (c10 * c22 - c12 * c20)
              + c02 * (c10 * c21 - c11 * c20);
    float inv_det = 1.0f / det;
    float i00 =  (c11 * c22 - c12 * c21) * inv_det;
    float i01 = -(c01 * c22 - c02 * c21) * inv_det;
    float i02 =  (c01 * c12 - c02 * c11) * inv_det;
    float i10 = -(c10 * c22 - c12 * c20) * inv_det;
    float i11 =  (c00 * c22 - c02 * c20) * inv_det;
    float i12 = -(c00 * c12 - c02 * c10) * inv_det;
    float i20 =  (c10 * c21 - c11 * c20) * inv_det;
    float i21 = -(c00 * c21 - c01 * c20) * inv_det;
    float i22 =  (c00 * c11 - c01 * c10) * inv_det;

    // ---- preload fractional coords f = pos @ inv(cell) in A-fragment layout ----
    for (int i = threadIdx.x; i < N_ATOMS; i += TPB) {
        float x = pos[i * 3 + 0], y = pos[i * 3 + 1], z = pos[i * 3 + 2];
        float fx = x * i00 + y * i10 + z * i20;
        float fy = x * i01 + y * i11 + z * i21;
        float fz = x * i02 + y * i12 + z * i22;
        v2f lo; lo.x = fx; lo.y = fy;
        v2f hi; hi.x = fz; hi.y = 0.0f;
        aLDS[i]           = lo;
        aLDS[N_ATOMS + i] = hi;
        qLDS[i] = q[i];
    }
    __syncthreads();

    const int wave = threadIdx.x >> 5;
    const int lane = threadIdx.x & 31;
    const int tile = blockIdx.x * WAVES_PER_BLOCK + wave;
    if (tile >= KTILES) return;          // wave-uniform exit: EXEC stays all-1s

    const int col  = lane & 15;          // k column owned by this lane
    const int half = lane >> 4;          // 0: rows 0-7 / K 0-1, 1: rows 8-15 / K 2-3

    // ---- this lane's k vector (n integers -> kvec via G = 2*pi*inv(cell)^T) ----
    int kidx = tile * 16 + col;
    bool valid = (kidx < KTOT);
    int kk = valid ? kidx : (KTOT - 1);
    int i1 = kk / (NKGRID * NKGRID);
    int rem = kk - i1 * (NKGRID * NKGRID);
    int i2 = rem / NKGRID;
    int i3 = rem - i2 * NKGRID;
    float n1 = (float)(i1 - 16);
    float n2 = (float)(i2 - 16);
    float n3 = (float)(i3 - 16);

    const float TWOPI = 6.28318530717958647692f;
    float kv0 = TWOPI * (n1 * i00 + n2 * i01 + n3 * i02);
    float kv1 = TWOPI * (n1 * i10 + n2 * i11 + n3 * i12);
    float kv2 = TWOPI * (n1 * i20 + n2 * i21 + n3 * i22);
    float k_sq = kv0 * kv0 + kv1 * kv1 + kv2 * kv2;

    // ---- B operand: 4x16 matrix of n-vector components (K=3 padded to 4) ----
    // B layout (32-bit, 4x16): lanes 0-15 hold K=0 (v0) / K=1 (v1),
    //                          lanes 16-31 hold K=2 (v0) / K=3 (v1)
    v2f b;
    b.x = half ? n3 : n1;
    b.y = half ? 0.0f : n2;

    // ---- atom loop: phase tile via WMMA, trig via normalized-input v_cos/v_sin ----
    float accRe = 0.0f, accIm = 0.0f;
    v8f zeroC = {};
    const v2f*   ap = &aLDS[half * N_ATOMS + col];
    const float* qp = &qLDS[half * 8];
    for (int t = 0; t < ATILES; ++t) {
        // single ds_load_b64: this lane's A fragment, already in register order
        v2f a = ap[t * 16];

        // D = A x B : 16 atoms x 16 k-vectors of phase (in revolutions)
        v8f d = __builtin_amdgcn_wmma_f32_16x16x4_f32(
            /*neg_a=*/false, a, /*neg_b=*/false, b,
            /*c_mod=*/(short)0, zeroC, /*reuse_a=*/false, /*reuse_b=*/false);

        // D layout: vgpr r, this lane -> atom M = r + 8*half, column N = col
#pragma unroll
        for (int r = 0; r < 8; ++r) {
            float qr = qp[t * 16 + r];
            // hw v_cos/v_sin take normalized input: cos(2*pi*x) == cos(kdr)
            accRe = __builtin_fmaf(qr, __builtin_amdgcn_cosf(d[r]), accRe);
            accIm = __builtin_fmaf(qr, __builtin_amdgcn_sinf(d[r]), accIm);
        }
    }

    // ---- fold the two half-wave row groups: lanes l and l+16 share column N ----
    accRe += __shfl_xor(accRe, 16, 32);
    accIm += __shfl_xor(accIm, 16, 32);
    float Ssq = accRe * accRe + accIm * accIm;

    // ---- kfac(k) = sum_j amp_j * exp(-k_sq * exp(2*shift_j)) ----
    float kfac = 0.0f;
#pragma unroll
    for (int j = 0; j < BW; ++j) {
        float mt = -__expf(2.0f * shift[j]);
        kfac = __builtin_fmaf(amp[j], __expf(k_sq * mt), kfac);
    }

    float v = (valid && (k_sq > 0.0f) && (half == 0)) ? (kfac * Ssq) : 0.0f;
#pragma unroll
    for (int off = 8; off >= 1; off >>= 1)
        v += __shfl_xor(v, off, 32);

    if (lane == 0)
        atomicAdd(out, v / det);
}

extern "C" void kernel_launch(void* const* d_in, const int* in_sizes, int n_in,
                              void* d_out, int out_size, void* d_ws, size_t ws_size,
                              hipStream_t stream) {
    const float* pos   = (const float*)d_in[0];
    const float* q     = (const float*)d_in[1];
    const float* cell  = (const float*)d_in[2];
    const float* shift = (const float*)d_in[3];
    const float* amp   = (const float*)d_in[4];
    const float* ene   = (const float*)d_in[5];
    // d_in[6] = batch (single system) -- unused
    float* out = (float*)d_out;

    hipLaunchKernelGGL(sog_init_kernel, dim3(1), dim3(1), 0, stream, out, ene);

    int blocks = (KTILES + WAVES_PER_BLOCK - 1) / WAVES_PER_BLOCK;
    hipLaunchKernelGGL(sog_main_kernel, dim3(blocks), dim3(TPB), 0, stream,
                       pos, q, cell, shift, amp, out);
    (void)in_sizes; (void)n_in; (void)out_size; (void)d_ws; (void)ws_size;
}